// RecurrentTemporalDeformableDecoder_75806172774975
// MI455X (gfx1250) — compile-verified
//
#include <hip/hip_runtime.h>
#include <hip/hip_bf16.h>
#include <math.h>

// ---------------------------------------------------------------------------
// RecurrentTemporalDeformableDecoder for MI455X (gfx1250, wave32, WMMA)
// B=8 T=32 Q=64 D=256 L=2 M=4 K=3 WL=7 GH=GW=24 FPS=30
// ---------------------------------------------------------------------------

#define BB 8
#define TT 32
#define QQ 64
#define DD 256
#define LL 2
#define MM 4
#define KK 3
#define WLEN 7
#define GHW 24
#define NSP 576      // GH*GW
#define ROWS 512     // B*Q
#define FFD 1024     // 4*D
#define APAD 36      // LDS row pitch (floats): 16B-aligned slots, conflict-free

typedef float v2f __attribute__((ext_vector_type(2)));
typedef float v8f __attribute__((ext_vector_type(8)));

#if defined(__gfx1250__) && __has_builtin(__builtin_amdgcn_global_load_async_to_lds_b128)
#define USE_ASYNC_LDS 1
#else
#define USE_ASYNC_LDS 0
#endif

__device__ __forceinline__ int imin_(int a, int b) { return a < b ? a : b; }

__device__ __forceinline__ float block_reduce_sum(float v, float* red) {
  const int tid = threadIdx.x;
  red[tid] = v;
  __syncthreads();
  for (int s = 128; s > 0; s >>= 1) {
    if (tid < s) red[tid] += red[tid + s];
    __syncthreads();
  }
  float r = red[0];
  __syncthreads();
  return r;
}

// ---------------------------------------------------------------------------
// Generic 32x32-tile FP32 WMMA GEMM:  C = act(A[M,K] @ W[K,N] + bias) + resid
// 128 threads = 4 waves, each wave owns one 16x16 WMMA tile.
// Tiles staged via CDNA5 async global->LDS DMA (ASYNCcnt) when available.
// act: 0 = none, 1 = exact GELU. All of M,N,K are multiples of 32 here.
// ---------------------------------------------------------------------------
__global__ __launch_bounds__(128) void gemm_wmma_kernel(
    const float* __restrict__ A, const float* __restrict__ W,
    const float* __restrict__ bias, const float* __restrict__ resid,
    float* __restrict__ C, int Mdim, int Ndim, int Kdim, int act) {
  __shared__ float As[32][APAD];
  __shared__ float Bs[32][APAD];
  const int tid = threadIdx.x;
  const int m0 = blockIdx.y * 32, n0 = blockIdx.x * 32;
  const int wv = tid >> 5, lane = tid & 31;
  const int ln = lane & 15, hi = lane >> 4;
  const int mt = (wv >> 1) << 4, nt = (wv & 1) << 4;
  // loader mapping: each thread moves 16B from rows lr0 and lr0+16
  const int lr0 = tid >> 3;          // 0..15
  const int lc0 = (tid & 7) << 2;    // 0,4,...,28
  v8f acc = {};
  for (int k0 = 0; k0 < Kdim; k0 += 32) {
#if USE_ASYNC_LDS
    {
      typedef int v4i_vs __attribute__((vector_size(16)));
      typedef __attribute__((address_space(1))) v4i_vs* gp_t;
      typedef __attribute__((address_space(3))) v4i_vs* lp_t;
      __builtin_amdgcn_global_load_async_to_lds_b128(
          (gp_t)&A[(m0 + lr0) * Kdim + k0 + lc0], (lp_t)&As[lr0][lc0], 0, 0);
      __builtin_amdgcn_global_load_async_to_lds_b128(
          (gp_t)&A[(m0 + lr0 + 16) * Kdim + k0 + lc0], (lp_t)&As[lr0 + 16][lc0], 0, 0);
      __builtin_amdgcn_global_load_async_to_lds_b128(
          (gp_t)&W[(k0 + lr0) * Ndim + n0 + lc0], (lp_t)&Bs[lr0][lc0], 0, 0);
      __builtin_amdgcn_global_load_async_to_lds_b128(
          (gp_t)&W[(k0 + lr0 + 16) * Ndim + n0 + lc0], (lp_t)&Bs[lr0 + 16][lc0], 0, 0);
#if __has_builtin(__builtin_amdgcn_s_wait_asynccnt)
      __builtin_amdgcn_s_wait_asynccnt(0);
#else
      asm volatile("s_wait_asynccnt 0" ::: "memory");
#endif
    }
#else
    {
      const float4 a0 = *(const float4*)&A[(m0 + lr0) * Kdim + k0 + lc0];
      const float4 a1 = *(const float4*)&A[(m0 + lr0 + 16) * Kdim + k0 + lc0];
      const float4 b0 = *(const float4*)&W[(k0 + lr0) * Ndim + n0 + lc0];
      const float4 b1 = *(const float4*)&W[(k0 + lr0 + 16) * Ndim + n0 + lc0];
      *(float4*)&As[lr0][lc0] = a0;
      *(float4*)&As[lr0 + 16][lc0] = a1;
      *(float4*)&Bs[lr0][lc0] = b0;
      *(float4*)&Bs[lr0 + 16][lc0] = b1;
    }
#endif
    __syncthreads();
#pragma unroll
    for (int kk = 0; kk < 32; kk += 4) {
      v2f a, bv;
      a.x = As[mt + ln][kk + 2 * hi];
      a.y = As[mt + ln][kk + 2 * hi + 1];
      bv.x = Bs[kk + 2 * hi][nt + ln];
      bv.y = Bs[kk + 2 * hi + 1][nt + ln];
#if __has_builtin(__builtin_amdgcn_wmma_f32_16x16x4_f32)
      acc = __builtin_amdgcn_wmma_f32_16x16x4_f32(
          /*neg_a=*/false, a, /*neg_b=*/false, bv,
          /*c_mod=*/(short)0, acc, /*reuse_a=*/false, /*reuse_b=*/false);
#else
#pragma unroll
      for (int r = 0; r < 8; ++r) acc[r] += a.x * bv.x + a.y * bv.y;
#endif
    }
    __syncthreads();
  }
#pragma unroll
  for (int r = 0; r < 8; ++r) {
    const int m = m0 + mt + r + 8 * hi;
    const int n = n0 + nt + ln;
    float v = acc[r];
    if (bias) v += bias[n];
    if (act == 1) v = 0.5f * v * (1.0f + erff(v * 0.70710678118654752f));
    if (resid) v += resid[m * Ndim + n];
    C[m * Ndim + n] = v;
  }
}

// ---------------------------------------------------------------------------
// Precompute kernels (run once; all tiny)
// ---------------------------------------------------------------------------

// out[r, :] = sincos_time(r*tscale + tbias, 256) @ W + bias   (one block per r)
__global__ __launch_bounds__(256) void sincos_proj_kernel(
    const float* __restrict__ W, const float* __restrict__ bias,
    float* __restrict__ outp, float tscale, float tbias) {
  __shared__ float sc[256];
  const int r = blockIdx.x, d = threadIdx.x;
  const float tv = (float)r * tscale + tbias;
  const float fr = expf(-logf(10000.0f) * (float)(d & 127) * (1.0f / 128.0f));
  const float ang = tv * fr;
  sc[d] = (d < 128) ? sinf(ang) : cosf(ang);
  __syncthreads();
  float acc = bias[d];
  for (int h = 0; h < 256; ++h) acc += sc[h] * W[h * 256 + d];
  outp[r * 256 + d] = acc;
}

__global__ __launch_bounds__(256) void pos2d_kernel(float* __restrict__ POS) {
  const int yx = blockIdx.x, d = threadIdx.x;
  const int y = yx / GHW, x = yx % GHW;
  const float fr = expf(-logf(10000.0f) * (float)(d & 63) * (1.0f / 64.0f));
  float v;
  if (d < 64)       v = sinf((float)y * fr);
  else if (d < 128) v = cosf((float)y * fr);
  else if (d < 192) v = sinf((float)x * fr);
  else              v = cosf((float)x * fr);
  POS[yx * 256 + d] = v;
}

// relh[i][w][c] = rel_feat[w] @ {delta|alpha}_w[i][256:512, c] + bias
__global__ __launch_bounds__(256) void relh_kernel(
    const float* __restrict__ RELF, const float* __restrict__ dw,
    const float* __restrict__ db, const float* __restrict__ aw,
    const float* __restrict__ ab, float* __restrict__ RELH) {
  const int e = threadIdx.x;
  if (e >= LL * WLEN * 16) return;
  const int i = e / (WLEN * 16);
  const int rm = e % (WLEN * 16);
  const int w = rm / 16, c = rm % 16;
  float acc = (c < 12) ? db[i * 12 + c] : ab[i * 4 + (c - 12)];
  for (int d = 0; d < 256; ++d) {
    const float rf = RELF[w * 256 + d];
    const float wt = (c < 12) ? dw[(i * 512 + 256 + d) * 12 + c]
                              : aw[(i * 512 + 256 + d) * 4 + (c - 12)];
    acc += rf * wt;
  }
  RELH[(i * WLEN + w) * 16 + c] = acc;
}

// Packed head weight [L][256][32]: cols 0-11 delta(qn part), 12-15 alpha, 16-17 refxy
__global__ __launch_bounds__(256) void whpack_kernel(
    const float* __restrict__ dw, const float* __restrict__ aw,
    const float* __restrict__ rw, float* __restrict__ WHPK) {
  const int e = blockIdx.x * blockDim.x + threadIdx.x;
  if (e >= LL * 256 * 32) return;
  const int i = e / (256 * 32);
  const int r = (e / 32) % 256;
  const int c = e % 32;
  float v = 0.0f;
  if (c < 12)      v = dw[(i * 512 + r) * 12 + c];
  else if (c < 16) v = aw[(i * 512 + r) * 4 + (c - 12)];
  else if (c < 18) v = rw[(i * 256 + r) * 2 + (c - 16)];
  WHPK[(i * 256 + r) * 32 + c] = v;
}

__global__ void hbpack_kernel(const float* __restrict__ rb, float* __restrict__ HB) {
  const int e = threadIdx.x;
  if (e >= LL * 32) return;
  const int i = e / 32, c = e % 32;
  HB[e] = (c >= 16 && c < 18) ? rb[i * 2 + (c - 16)] : 0.0f;
}

__global__ void qinitm_kernel(const float* __restrict__ initq,
                              const float* __restrict__ miw,
                              const float* __restrict__ mib,
                              float* __restrict__ outq) {
  const int q = threadIdx.x;
  if (q >= QQ) return;
  float acc = mib[0];
  for (int d = 0; d < 256; ++d) acc += initq[q * 256 + d] * miw[d];
  outq[q] = acc;
}

__global__ __launch_bounds__(256) void previnit_kernel(
    const float* __restrict__ initq, float* __restrict__ PREV) {
  const int e = blockIdx.x * blockDim.x + threadIdx.x;
  if (e >= ROWS * DD) return;
  const int q = (e >> 8) & 63;
  PREV[e] = initq[q * 256 + (e & 255)];
}

// ---------------------------------------------------------------------------
// Per-timestep kernels
// ---------------------------------------------------------------------------

// Query mix + LN. One block per (b,q) row.
__global__ __launch_bounds__(256) void mix_kernel(
    const float* __restrict__ PREV, const float* __restrict__ initq,
    const float* __restrict__ QINITM, const float* __restrict__ mpw,
    const float* __restrict__ mpb, const float* __restrict__ CTXM,
    const float* __restrict__ TIMEM, const float* __restrict__ mlg,
    const float* __restrict__ mlb, float* __restrict__ QCUR, int t) {
  __shared__ float red[256];
  const int row = blockIdx.x, tid = threadIdx.x;
  const int b = row >> 6, q = row & 63;
  const float p = PREV[row * 256 + tid];
  const float iq = initq[q * 256 + tid];
  const float s_prev = block_reduce_sum(p * mpw[tid], red) + mpb[0];
  const float s_init = QINITM[q];
  const float mx = fmaxf(s_prev, s_init);
  const float e0 = expf(s_prev - mx), e1 = expf(s_init - mx);
  const float inv = 1.0f / (e0 + e1);
  const float v = (e0 * inv) * p + (e1 * inv) * iq +
                  CTXM[(b * TT + t) * 256 + tid] + TIMEM[t * 256 + tid];
  const float mean = block_reduce_sum(v, red) * (1.0f / 256.0f);
  const float dv = v - mean;
  const float var = block_reduce_sum(dv * dv, red) * (1.0f / 256.0f);
  QCUR[row * 256 + tid] = dv * rsqrtf(var + 1e-5f) * mlg[tid] + mlb[tid];
}

// Row-wise LayerNorm over 256 channels. One block per row.
__global__ __launch_bounds__(256) void ln_kernel(
    const float* __restrict__ src, float* __restrict__ dst,
    const float* __restrict__ g, const float* __restrict__ b) {
  __shared__ float red[256];
  const int row = blockIdx.x, tid = threadIdx.x;
  const float v = src[row * 256 + tid];
  const float mean = block_reduce_sum(v, red) * (1.0f / 256.0f);
  const float dv = v - mean;
  const float var = block_reduce_sum(dv * dv, red) * (1.0f / 256.0f);
  dst[row * 256 + tid] = dv * rsqrtf(var + 1e-5f) * g[tid] + b[tid];
}

// Deformable trilinear sampling + softmax aggregation. One block per (b,q).
// kv is never materialized: kv = patch + pos2d + absw (masked by tau in [0,T)).
__global__ __launch_bounds__(256) void sample_kernel(
    const float* __restrict__ HD, const float* __restrict__ RELH,
    const float* __restrict__ patch, const float* __restrict__ POS2D,
    const float* __restrict__ ABSW, const float* __restrict__ lamp,
    float* __restrict__ AGG, int t, int layer) {
  __shared__ float s_wt[28][8];
  __shared__ int s_tau[28][8];
  __shared__ int s_sp[28][4];
  __shared__ float s_logit[28];
  __shared__ float s_wgt[28];
  const int row = blockIdx.x, tid = threadIdx.x;
  const int b = row >> 6;

  if (tid < 28) {
    const int w = tid >> 2, m = tid & 3;
    const float* hd = HD + row * 32;
    const float* rh = RELH + (layer * WLEN + w) * 16;
    const float dx = tanhf(hd[m * 3 + 0] + rh[m * 3 + 0]);
    const float dy = tanhf(hd[m * 3 + 1] + rh[m * 3 + 1]);
    const float dt = tanhf(hd[m * 3 + 2] + rh[m * 3 + 2]);
    const float logit = hd[12 + m] + rh[12 + m];
    const float refx = 1.0f / (1.0f + expf(-hd[16]));
    const float refy = 1.0f / (1.0f + expf(-hd[17]));
    float cx = fminf(fmaxf(refx + dx, 0.0f), 1.0f) * (float)GHW - 0.5f;
    float cy = fminf(fmaxf(refy + dy, 0.0f), 1.0f) * (float)GHW - 0.5f;
    cx = fminf(fmaxf(cx, 0.0f), (float)(GHW - 1));
    cy = fminf(fmaxf(cy, 0.0f), (float)(GHW - 1));
    const float x0f = floorf(cx), y0f = floorf(cy);
    const float wx = cx - x0f, wy = cy - y0f;
    const int x0 = (int)x0f, y0 = (int)y0f;
    const int x1 = imin_(x0 + 1, GHW - 1), y1 = imin_(y0 + 1, GHW - 1);
    // temporal (unclamped alpha -> extrapolation, matching reference)
    const float target = (float)w + dt;
    const float t0f = fminf(fmaxf(floorf(target), 0.0f), (float)(WLEN - 1));
    const int t0i = (int)t0f;
    const int t1i = imin_(t0i + 1, WLEN - 1);
    const float alpha = target - t0f;
    int tau0 = t + t0i - KK; if (tau0 < 0 || tau0 >= TT) tau0 = -1;
    int tau1 = t + t1i - KK; if (tau1 < 0 || tau1 >= TT) tau1 = -1;
    const float lam = log1pf(expf(lamp[layer]));
    const float absd = fabsf((float)w - (float)KK) * (1.0f / 30.0f);
    s_logit[tid] = logit - lam * absd;
    const float w00 = (1 - wx) * (1 - wy), w01 = wx * (1 - wy);
    const float w10 = (1 - wx) * wy, w11 = wx * wy;
    const float a0 = 1.0f - alpha;
    s_wt[tid][0] = a0 * w00; s_wt[tid][1] = a0 * w01;
    s_wt[tid][2] = a0 * w10; s_wt[tid][3] = a0 * w11;
    s_wt[tid][4] = alpha * w00; s_wt[tid][5] = alpha * w01;
    s_wt[tid][6] = alpha * w10; s_wt[tid][7] = alpha * w11;
    s_sp[tid][0] = y0 * GHW + x0; s_sp[tid][1] = y0 * GHW + x1;
    s_sp[tid][2] = y1 * GHW + x0; s_sp[tid][3] = y1 * GHW + x1;
#pragma unroll
    for (int c = 0; c < 4; ++c) { s_tau[tid][c] = tau0; s_tau[tid][4 + c] = tau1; }
  }
  __syncthreads();
  if (tid < WLEN) {  // softmax over m within each window slot
    float mx = -1e30f;
    for (int m = 0; m < 4; ++m) mx = fmaxf(mx, s_logit[tid * 4 + m]);
    float e[4], sum = 0.0f;
    for (int m = 0; m < 4; ++m) { e[m] = expf(s_logit[tid * 4 + m] - mx); sum += e[m]; }
    const float inv = 1.0f / sum;
    for (int m = 0; m < 4; ++m) s_wgt[tid * 4 + m] = e[m] * inv;
  }
  __syncthreads();
  if (tid < 28) {
    const float g = s_wgt[tid];
#pragma unroll
    for (int c = 0; c < 8; ++c) s_wt[tid][c] *= g;
  }
  __syncthreads();

  const int d = tid;
  float acc = 0.0f;
  for (int j = 0; j < 28; ++j) {
#pragma unroll
    for (int c = 0; c < 8; ++c) {
      const int tau = s_tau[j][c];
      if (tau >= 0) {
        const int sp = s_sp[j][c & 3];
        acc += s_wt[j][c] * (patch[((b * TT + tau) * NSP + sp) * 256 + d] +
                             POS2D[sp * 256 + d] + ABSW[tau * 256 + d]);
      }
    }
  }
  AGG[row * 256 + d] = acc;
}

__global__ __launch_bounds__(256) void writeout_kernel(
    const float* __restrict__ QCUR, float* __restrict__ PREV,
    float* __restrict__ outp, int t) {
  const int e = blockIdx.x * blockDim.x + threadIdx.x;
  if (e >= ROWS * DD) return;
  const float v = QCUR[e];
  PREV[e] = v;
  const int d = e & 255, bq = e >> 8;
  const int q = bq & 63, b = bq >> 6;
  outp[(((b * TT + t) * QQ) + q) * 256 + d] = v;  // [B,T,Q,D]
}

// ---------------------------------------------------------------------------
// Host orchestration
// ---------------------------------------------------------------------------
static inline void launch_gemm(const float* A, const float* W, const float* bias,
                               const float* resid, float* C, int M, int N, int Kd,
                               int act, hipStream_t stream) {
  dim3 grid(N / 32, M / 32);
  gemm_wmma_kernel<<<grid, 128, 0, stream>>>(A, W, bias, resid, C, M, N, Kd, act);
}

extern "C" void kernel_launch(void* const* d_in, const int* in_sizes, int n_in,
                              void* d_out, int out_size, void* d_ws, size_t ws_size,
                              hipStream_t stream) {
  (void)in_sizes; (void)n_in; (void)out_size; (void)ws_size;
  const float* patch = (const float*)d_in[0];
  const float* tok   = (const float*)d_in[1];
  const float* initq = (const float*)d_in[2];
  const float* ctxw  = (const float*)d_in[3];
  const float* ctxb  = (const float*)d_in[4];
  const float* apw   = (const float*)d_in[5];
  const float* apb   = (const float*)d_in[6];
  const float* rpw   = (const float*)d_in[7];
  const float* rpb   = (const float*)d_in[8];
  const float* mpw   = (const float*)d_in[9];
  const float* mpb   = (const float*)d_in[10];
  const float* miw   = (const float*)d_in[11];
  const float* mib   = (const float*)d_in[12];
  const float* mcw   = (const float*)d_in[13];
  const float* mcb   = (const float*)d_in[14];
  const float* mtw   = (const float*)d_in[15];
  const float* mtb   = (const float*)d_in[16];
  const float* mlg   = (const float*)d_in[17];
  const float* mlb   = (const float*)d_in[18];
  const float* rxw   = (const float*)d_in[19];
  const float* rxb   = (const float*)d_in[20];
  const float* dlw   = (const float*)d_in[21];
  const float* dlb   = (const float*)d_in[22];
  const float* alw   = (const float*)d_in[23];
  const float* alb   = (const float*)d_in[24];
  const float* lam   = (const float*)d_in[25];
  const float* pjw   = (const float*)d_in[26];
  const float* pjb   = (const float*)d_in[27];
  const float* n1g   = (const float*)d_in[28];
  const float* n1b   = (const float*)d_in[29];
  const float* n2g   = (const float*)d_in[30];
  const float* n2b   = (const float*)d_in[31];
  const float* f1w   = (const float*)d_in[32];
  const float* f1b   = (const float*)d_in[33];
  const float* f2w   = (const float*)d_in[34];
  const float* f2b   = (const float*)d_in[35];
  float* outp = (float*)d_out;

  // workspace carve-up (floats)
  float* ws    = (float*)d_ws;
  float* POS2D = ws;                    // 576*256
  float* ABSW  = POS2D + NSP * DD;      // 32*256
  float* RELF  = ABSW + TT * DD;        // 7*256
  float* RELH  = RELF + WLEN * DD;      // 2*7*16
  float* WHPK  = RELH + LL * WLEN * 16; // 2*256*32
  float* HBPK  = WHPK + LL * DD * 32;   // 2*32
  float* QIM   = HBPK + LL * 32;        // 64
  float* CT0   = QIM + QQ;              // 256*256
  float* CTXM  = CT0 + BB * TT * DD;    // 256*256
  float* TIMEM = CTXM + BB * TT * DD;   // 32*256
  float* PREV  = TIMEM + TT * DD;       // 512*256
  float* QCUR  = PREV + ROWS * DD;
  float* QN    = QCUR + ROWS * DD;
  float* HD    = QN + ROWS * DD;        // 512*32
  float* AGG   = HD + ROWS * 32;        // 512*256
  float* OUTB  = AGG + ROWS * DD;
  float* HBUF  = OUTB + ROWS * DD;
  float* GBUF  = HBUF + ROWS * DD;      // 512*1024

  // ---- one-time precompute (time-invariant across the scan) ----
  pos2d_kernel<<<NSP, 256, 0, stream>>>(POS2D);
  // absw[t] = sincos(t/FPS) @ abs_proj + b  (serves e_abs AND abs_tau)
  sincos_proj_kernel<<<TT, 256, 0, stream>>>(apw, apb, ABSW, 1.0f / 30.0f, 0.0f);
  // rel_feat[w] = sincos((w-K)/FPS) @ rel_proj + b
  sincos_proj_kernel<<<WLEN, 256, 0, stream>>>(rpw, rpb, RELF, 1.0f / 30.0f, -3.0f / 30.0f);
  relh_kernel<<<1, 256, 0, stream>>>(RELF, dlw, dlb, alw, alb, RELH);
  whpack_kernel<<<(LL * 256 * 32 + 255) / 256, 256, 0, stream>>>(dlw, alw, rxw, WHPK);
  hbpack_kernel<<<1, 64, 0, stream>>>(rxb, HBPK);
  qinitm_kernel<<<1, 64, 0, stream>>>(initq, miw, mib, QIM);
  launch_gemm(tok, ctxw, ctxb, nullptr, CT0, BB * TT, DD, DD, 0, stream);
  launch_gemm(CT0, mcw, mcb, nullptr, CTXM, BB * TT, DD, DD, 0, stream);
  launch_gemm(ABSW, mtw, mtb, nullptr, TIMEM, TT, DD, DD, 0, stream);
  previnit_kernel<<<(ROWS * DD + 255) / 256, 256, 0, stream>>>(initq, PREV);

  // ---- sequential scan over T ----
  for (int t = 0; t < TT; ++t) {
    mix_kernel<<<ROWS, 256, 0, stream>>>(PREV, initq, QIM, mpw, mpb, CTXM, TIMEM,
                                         mlg, mlb, QCUR, t);
    for (int i = 0; i < LL; ++i) {
      ln_kernel<<<ROWS, 256, 0, stream>>>(QCUR, QN, n1g + i * DD, n1b + i * DD);
      // heads: delta(12) | alpha(4) | refxy(2) in one 512x32 GEMM
      launch_gemm(QN, WHPK + i * DD * 32, HBPK + i * 32, nullptr, HD,
                  ROWS, 32, DD, 0, stream);
      sample_kernel<<<ROWS, 256, 0, stream>>>(HD, RELH, patch, POS2D, ABSW, lam,
                                              AGG, t, i);
      launch_gemm(AGG, pjw + i * DD * DD, pjb + i * DD, QCUR, OUTB,
                  ROWS, DD, DD, 0, stream);
      ln_kernel<<<ROWS, 256, 0, stream>>>(OUTB, HBUF, n2g + i * DD, n2b + i * DD);
      launch_gemm(HBUF, f1w + i * DD * FFD, f1b + i * FFD, nullptr, GBUF,
                  ROWS, FFD, DD, 1, stream);  // fused exact GELU
      launch_gemm(GBUF, f2w + i * FFD * DD, f2b + i * DD, OUTB, QCUR,
                  ROWS, DD, FFD, 0, stream);
    }
    writeout_kernel<<<(ROWS * DD + 255) / 256, 256, 0, stream>>>(QCUR, PREV, outp, t);
  }
}